// Q_U1_control_46789373723419
// MI455X (gfx1250) — compile-verified
//
#include <hip/hip_runtime.h>

// out[b,x,y,i] = arccos(clip(sum_c cos(pi * in[b,x,y,c]), -1+eps, 1-eps)) / pi
// (quantum circuit collapses analytically: qubit 0 is only ever a control, so
//  <Z0> = cos(pi * a0) independent of the gate exponents / other angles)

typedef __attribute__((ext_vector_type(2))) float v2f;
typedef __attribute__((ext_vector_type(8))) float v8f;

namespace {
constexpr int BATCH = 16;
constexpr int H = 14;
constexpr int W = 14;
constexpr int C = 3;
constexpr int NX = 12;
constexpr int NY = 12;
constexpr int NF = 4;
constexpr int ROWS = BATCH * NX * NY;          // 2304 output positions
constexpr int TILES = ROWS / 16;               // 144 wave-tiles (16 rows each)
constexpr int WAVES_PER_BLOCK = 4;             // 128 threads
constexpr int BLOCKS = TILES / WAVES_PER_BLOCK;// 36 blocks
constexpr float CLIP_EPS = 1e-5f;
constexpr float INV_PI = 0.31830988618379067154f;
}

__global__ __launch_bounds__(128) void quanv_wmma_kernel(
    const float* __restrict__ in, float* __restrict__ out) {
  const int lane = threadIdx.x & 31;
  const int tile = blockIdx.x * WAVES_PER_BLOCK + (threadIdx.x >> 5);

  // ---------------- load phase: build A-matrix (16x4 f32) ----------------
  // lanes 0-15 : row M = lane,      K-slots = (cospi(c0), cospi(c1))
  // lanes 16-31: row M = lane - 16, K-slots = (cospi(c2), 0)
  const int mload = lane & 15;
  const int rload = tile * 16 + mload;                 // global row (b,x,y)
  const int b   = rload / (NX * NY);
  const int rem = rload - b * (NX * NY);
  const int x   = rem / NY;
  const int y   = rem - x * NY;
  const int base = ((b * H + x) * W + y) * C;          // top-left corner of patch
  const bool lo = (lane < 16);

  const float e0 = in[base + (lo ? 0 : 2)];
  const float e1 = in[base + 1];                       // in-bounds for all lanes

  v2f amat;
  amat.x = cospif(e0);
  amat.y = lo ? cospif(e1) : 0.0f;

  // ---------------- store-phase row mapping ----------------
  const int n = lane & 15;
  const int h = lane >> 4;
  const int mstore = h * 8 + (n & 7);
  const int rstore = tile * 16 + mstore;

  float sum;
#if __has_builtin(__builtin_amdgcn_wmma_f32_16x16x4_f32)
  // B = all-ones (4x16): row sum lands in every column of D.
  v2f bmat;
  bmat.x = 1.0f;
  bmat.y = 1.0f;
  v8f acc = {};
  // D = A x B + 0  ->  D[M, N] = sum_k A[M, k]
  acc = __builtin_amdgcn_wmma_f32_16x16x4_f32(
      /*neg_a=*/false, amat, /*neg_b=*/false, bmat,
      /*c_mod=*/(short)0, acc, /*reuse_a=*/false, /*reuse_b=*/false);
  // D layout: VGPR j, lane half h -> row M = 8h + j (any column).
  {
    const int j = n & 7;
    sum = (j & 4) ? ((j & 2) ? ((j & 1) ? acc.s7 : acc.s6)
                             : ((j & 1) ? acc.s5 : acc.s4))
                  : ((j & 2) ? ((j & 1) ? acc.s3 : acc.s2)
                             : ((j & 1) ? acc.s1 : acc.s0));
  }
#else
  // scalar fallback: recompute the 3-channel sum for the store row directly
  {
    const int b2   = rstore / (NX * NY);
    const int rem2 = rstore - b2 * (NX * NY);
    const int x2   = rem2 / NY;
    const int y2   = rem2 - x2 * NY;
    const int base2 = ((b2 * H + x2) * W + y2) * C;
    sum = cospif(in[base2 + 0]) + cospif(in[base2 + 1]) + cospif(in[base2 + 2]);
  }
  (void)amat;
#endif

  const float t = fminf(fmaxf(sum, -1.0f + CLIP_EPS), 1.0f - CLIP_EPS);
  const float r = acosf(t) * INV_PI;

  // Each output row needs NF=4 identical values; the two lanes that share a
  // row (n and n+8 within a half) write filter pairs {0,1} and {2,3}.
  const int fofs = (n < 8) ? 0 : 2;
  float2 o;
  o.x = r;
  o.y = r;
  *reinterpret_cast<float2*>(out + rstore * NF + fofs) = o;
}

extern "C" void kernel_launch(void* const* d_in, const int* in_sizes, int n_in,
                              void* d_out, int out_size, void* d_ws, size_t ws_size,
                              hipStream_t stream) {
  (void)in_sizes; (void)n_in; (void)d_ws; (void)ws_size; (void)out_size;
  const float* in = (const float*)d_in[0];
  // d_in[1] (gate exponents) is provably dead: <Z0> is independent of it.
  float* out = (float*)d_out;
  quanv_wmma_kernel<<<dim3(BLOCKS), dim3(WAVES_PER_BLOCK * 32), 0, stream>>>(in, out);
}